// SOG_3DmixedSpecies_Water_309237645346
// MI455X (gfx1250) — compile-verified
//
#include <hip/hip_runtime.h>
#include <hip/hip_bf16.h>
#include <math.h>

typedef float v2f __attribute__((ext_vector_type(2)));
typedef float v8f __attribute__((ext_vector_type(8)));

// ---------------- problem constants ----------------
constexpr int NM    = 31;        // modes per dim
constexpr int NP    = 961;       // 31*31
constexpr int NPPAD = 992;       // W rows padded to mult of 32 (N-dim of nufft1 GEMM)
constexpr int KNP   = 964;       // np padded to mult of 4 (K-dim of stage-A GEMM)
constexpr int J     = 768;       // particles per batch
constexpr int JOs   = 256;       // oxygen count (cols 0..255); H = cols 256..767
constexpr int CCH   = 4;         // charge channels
constexpr int CMP   = 128;       // C*31=124 padded to 128 (M-dim of nufft1)
constexpr int QMR   = 496;       // 16*31 valid rows of g/T
constexpr int QMP   = 512;       // padded to mult of 32 (M-dim of stage A)
constexpr float LBOX   = 20.0f;
constexpr float VBOX   = 8000.0f;
constexpr float TWOPI_L = 0.31415926535897931f;  // 2*pi/L

// ---------------- workspace layout (floats) ----------------
constexpr size_t SZ_PH  = (size_t)NM * J;
constexpr size_t OFF_CX = 0;
constexpr size_t OFF_SX = OFF_CX + SZ_PH;
constexpr size_t OFF_CY = OFF_SX + SZ_PH;
constexpr size_t OFF_SY = OFF_CY + SZ_PH;
constexpr size_t OFF_CZ = OFF_SY + SZ_PH;
constexpr size_t OFF_SZ = OFF_CZ + SZ_PH;
constexpr size_t SZ_W   = (size_t)NPPAD * J;
constexpr size_t OFF_WC = OFF_SZ + SZ_PH;
constexpr size_t OFF_WS = OFF_WC + SZ_W;
constexpr size_t SZ_Y   = (size_t)CMP * J;
constexpr size_t OFF_YR = OFF_WS + SZ_W;
constexpr size_t OFF_YI = OFF_YR + SZ_Y;
constexpr size_t SZ_R   = (size_t)CMP * NPPAD;
constexpr size_t OFF_ROR = OFF_YI + SZ_Y;
constexpr size_t OFF_ROI = OFF_ROR + SZ_R;
constexpr size_t OFF_RHR = OFF_ROI + SZ_R;
constexpr size_t OFF_RHI = OFF_RHR + SZ_R;
constexpr size_t SZ_G   = (size_t)QMP * KNP;
constexpr size_t OFF_GOR = OFF_RHI + SZ_R;
constexpr size_t OFF_GOI = OFF_GOR + SZ_G;
constexpr size_t OFF_GHR = OFF_GOI + SZ_G;
constexpr size_t OFF_GHI = OFF_GHR + SZ_G;
constexpr size_t SZ_T   = (size_t)QMP * J;
constexpr size_t OFF_TR = OFF_GHI + SZ_G;
constexpr size_t OFF_TI = OFF_TR + SZ_T;
constexpr size_t WS_TOTAL = OFF_TI + SZ_T;   // ~5.1M floats ~= 20.5 MB

// ---------------- phase tables: C/S[d][m][j] = cos/sin((m-15)*theta_j^d) ----------------
__global__ void phase_build(const float* __restrict__ pos,
                            float* __restrict__ Cx, float* __restrict__ Sx,
                            float* __restrict__ Cy, float* __restrict__ Sy,
                            float* __restrict__ Cz, float* __restrict__ Sz) {
  int t = blockIdx.x * blockDim.x + threadIdx.x;
  if (t >= 3 * J) return;
  int d = t / J, j = t % J;
  float th = TWOPI_L * (pos[j * 3 + d] - 0.5f * LBOX);
  float* C = (d == 0) ? Cx : (d == 1) ? Cy : Cz;
  float* S = (d == 0) ? Sx : (d == 1) ? Sy : Sz;
  for (int mi = 0; mi < NM; ++mi) {
    float s, c;
    sincosf((float)(mi - 15) * th, &s, &c);
    C[mi * J + j] = c;
    S[mi * J + j] = s;
  }
}

// W[np][j] = e^{i(n y + p z)} split into cos (Wc) and sin (Ws); rows 961..991 stay zero
__global__ void w_build(const float* __restrict__ Cy, const float* __restrict__ Sy,
                        const float* __restrict__ Cz, const float* __restrict__ Sz,
                        float* __restrict__ Wc, float* __restrict__ Ws) {
  int t = blockIdx.x * blockDim.x + threadIdx.x;
  if (t >= NP * J) return;
  int np = t / J, j = t % J;
  int n = np / NM, p = np % NM;
  float cy = Cy[n * J + j], sy = Sy[n * J + j];
  float cz = Cz[p * J + j], sz = Sz[p * J + j];
  Wc[(size_t)np * J + j] = cy * cz - sy * sz;
  Ws[(size_t)np * J + j] = sy * cz + cy * sz;
}

// Y[(c,m)][j] = charge[j][c] * e^{-i m x_j}; rows 124..127 stay zero
__global__ void y_build(const float* __restrict__ charge,
                        const float* __restrict__ Cx, const float* __restrict__ Sx,
                        float* __restrict__ Yr, float* __restrict__ Yi) {
  int t = blockIdx.x * blockDim.x + threadIdx.x;
  if (t >= (CCH * NM) * J) return;
  int cm = t / J, j = t % J;
  int c = cm / NM, m = cm % NM;
  float q = charge[j * CCH + c];
  Yr[(size_t)cm * J + j] =  q * Cx[m * J + j];
  Yi[(size_t)cm * J + j] = -q * Sx[m * J + j];
}

// -------- complex WMMA GEMM, 2x2 register-blocked (32x32 complex per wave) --------
// NTMODE: B stored [N][K] (K-contiguous, float2 loads);  else B stored [K][N].
// SPLUS:  s=+1 (stage A, conj phases)  else s=-1 (nufft1).
// Per K-step: 8 x b64 loads -> 16 v_wmma_f32_16x16x4_f32 (2:1 wmma:vmem).
template <bool NTMODE, bool SPLUS>
__global__ __launch_bounds__(32)
void cgemm_wmma(const float* __restrict__ Ar, const float* __restrict__ Ai, int lda,
                const float* __restrict__ Br, const float* __restrict__ Bi, int ldb,
                float* __restrict__ Cr, float* __restrict__ Ci, int ldc, int K) {
  const int lane = threadIdx.x;
  const int row  = lane & 15;   // M (A) / N (B,C) row within 16x16 tile
  const int kp   = lane >> 4;   // K-pair selector
  const int n0 = blockIdx.x * 32;
  const int m0 = blockIdx.y * 32;

  v8f accr[2][2], acci[2][2];
#pragma unroll
  for (int mt = 0; mt < 2; ++mt)
#pragma unroll
    for (int nt = 0; nt < 2; ++nt) {
      accr[mt][nt] = (v8f){0.f, 0.f, 0.f, 0.f, 0.f, 0.f, 0.f, 0.f};
      acci[mt][nt] = (v8f){0.f, 0.f, 0.f, 0.f, 0.f, 0.f, 0.f, 0.f};
    }

  const float* arp0 = Ar + (size_t)(m0 + row) * lda + 2 * kp;
  const float* aip0 = Ai + (size_t)(m0 + row) * lda + 2 * kp;
  const float* arp1 = arp0 + (size_t)16 * lda;
  const float* aip1 = aip0 + (size_t)16 * lda;

  for (int k0 = 0; k0 < K; k0 += 4) {
    v2f a_r[2], a_i[2], b_r[2], b_i[2];
    a_r[0] = *(const v2f*)(arp0 + k0);
    a_r[1] = *(const v2f*)(arp1 + k0);
    a_i[0] = *(const v2f*)(aip0 + k0);
    a_i[1] = *(const v2f*)(aip1 + k0);
    if (NTMODE) {
      const size_t bo = (size_t)(n0 + row) * ldb + k0 + 2 * kp;
      b_r[0] = *(const v2f*)(Br + bo);
      b_i[0] = *(const v2f*)(Bi + bo);
      b_r[1] = *(const v2f*)(Br + bo + (size_t)16 * ldb);
      b_i[1] = *(const v2f*)(Bi + bo + (size_t)16 * ldb);
    } else {
      const int kk = k0 + 2 * kp;
#pragma unroll
      for (int nt = 0; nt < 2; ++nt) {
        const size_t c0 = (size_t)kk * ldb + n0 + nt * 16 + row;
        b_r[nt].x = Br[c0];
        b_r[nt].y = Br[c0 + ldb];
        b_i[nt].x = Bi[c0];
        b_i[nt].y = Bi[c0 + ldb];
      }
    }
    // negated tile per M-block: s=+1 needs -ai (for Cr), s=-1 needs -ar (for Ci)
    v2f neg[2];
#pragma unroll
    for (int mt = 0; mt < 2; ++mt) {
      v2f src = SPLUS ? a_i[mt] : a_r[mt];
      neg[mt].x = -src.x;
      neg[mt].y = -src.y;
    }
    // (ar + i ai)(br + i s bi): Cr += ar*br - s*ai*bi ; Ci += ai*br + s*ar*bi
#pragma unroll
    for (int mt = 0; mt < 2; ++mt)
#pragma unroll
      for (int nt = 0; nt < 2; ++nt) {
        accr[mt][nt] = __builtin_amdgcn_wmma_f32_16x16x4_f32(
            false, a_r[mt], false, b_r[nt], (short)0, accr[mt][nt], false, false);
        acci[mt][nt] = __builtin_amdgcn_wmma_f32_16x16x4_f32(
            false, a_i[mt], false, b_r[nt], (short)0, acci[mt][nt], false, false);
        if (SPLUS) {
          accr[mt][nt] = __builtin_amdgcn_wmma_f32_16x16x4_f32(
              false, neg[mt], false, b_i[nt], (short)0, accr[mt][nt], false, false);
          acci[mt][nt] = __builtin_amdgcn_wmma_f32_16x16x4_f32(
              false, a_r[mt], false, b_i[nt], (short)0, acci[mt][nt], false, false);
        } else {
          accr[mt][nt] = __builtin_amdgcn_wmma_f32_16x16x4_f32(
              false, a_i[mt], false, b_i[nt], (short)0, accr[mt][nt], false, false);
          acci[mt][nt] = __builtin_amdgcn_wmma_f32_16x16x4_f32(
              false, neg[mt], false, b_i[nt], (short)0, acci[mt][nt], false, false);
        }
      }
  }

#pragma unroll
  for (int mt = 0; mt < 2; ++mt)
#pragma unroll
    for (int nt = 0; nt < 2; ++nt)
#pragma unroll
      for (int v = 0; v < 8; ++v) {
        const size_t o = (size_t)(m0 + mt * 16 + v + 8 * kp) * ldc + n0 + nt * 16 + row;
        Cr[o] = accr[mt][nt][v];
        Ci[o] = acci[mt][nt][v];
      }
}

// Green's-function mixing + gradient channels:
// g[(comp*4+c)*31+m][np] = {1,kx,ky,kz} * (m_SS' * r) for each species
__global__ void g_build(const float* __restrict__ rOr, const float* __restrict__ rOi,
                        const float* __restrict__ rHr, const float* __restrict__ rHi,
                        const float* __restrict__ shOO, const float* __restrict__ amOO,
                        const float* __restrict__ shOH, const float* __restrict__ amOH,
                        const float* __restrict__ shHO, const float* __restrict__ amHO,
                        const float* __restrict__ shHH, const float* __restrict__ amHH,
                        float* __restrict__ gOr, float* __restrict__ gOi,
                        float* __restrict__ gHr, float* __restrict__ gHi) {
  int t = blockIdx.x * blockDim.x + threadIdx.x;
  if (t >= NM * NP) return;
  int m = t / NP, np = t % NP;
  int n = np / NM, p = np % NM;
  float kx = TWOPI_L * (float)(m - 15);
  float ky = TWOPI_L * (float)(n - 15);
  float kz = TWOPI_L * (float)(p - 15);
  float squ = kx * kx + ky * ky + kz * kz;

  float mOO = 0.f, mOH = 0.f, mHO = 0.f, mHH = 0.f;
#pragma unroll
  for (int i = 0; i < 6; ++i) {
    mOO += amOO[i] * expf(-squ * expf(2.0f * shOO[i]));
    mOH += amOH[i] * expf(-squ * expf(2.0f * shOH[i]));
    mHO += amHO[i] * expf(-squ * expf(2.0f * shHO[i]));
    mHH += amHH[i] * expf(-squ * expf(2.0f * shHH[i]));
  }
  if (squ == 0.0f) { mOH = 0.f; mHO = 0.f; mHH = 0.f; }  // k=0 masked (OO kept)

#pragma unroll
  for (int c = 0; c < CCH; ++c) {
    const size_t ri = (size_t)(c * NM + m) * NPPAD + np;
    float ror = rOr[ri], roi = rOi[ri];
    float rhr = rHr[ri], rhi = rHi[ri];
    float fOr = mOO * ror + mOH * rhr, fOi = mOO * roi + mOH * rhi;
    float fHr = mHO * ror + mHH * rhr, fHi = mHO * roi + mHH * rhi;
    const size_t r0 = (size_t)(c * NM + m)        * KNP + np;
    const size_t r1 = (size_t)((4  + c) * NM + m) * KNP + np;
    const size_t r2 = (size_t)((8  + c) * NM + m) * KNP + np;
    const size_t r3 = (size_t)((12 + c) * NM + m) * KNP + np;
    gOr[r0] = fOr;        gOi[r0] = fOi;
    gOr[r1] = kx * fOr;   gOi[r1] = kx * fOi;
    gOr[r2] = ky * fOr;   gOi[r2] = ky * fOi;
    gOr[r3] = kz * fOr;   gOi[r3] = kz * fOi;
    gHr[r0] = fHr;        gHi[r0] = fHi;
    gHr[r1] = kx * fHr;   gHi[r1] = kx * fHi;
    gHr[r2] = ky * fHr;   gHi[r2] = ky * fHi;
    gHr[r3] = kz * fHr;   gHi[r3] = kz * fHi;
  }
}

// u[q,j] = sum_m T[q*31+m,j] * e^{+i m x_j}; energy = Re(u[c])/2V, force = Im(u[4..15])/V
__global__ void finalize(const float* __restrict__ Tr, const float* __restrict__ Ti,
                         const float* __restrict__ Cx, const float* __restrict__ Sx,
                         const float* __restrict__ charge, float* __restrict__ out, int b) {
  int j = blockIdx.x * blockDim.x + threadIdx.x;
  if (j >= J) return;
  float ur[4]  = {0.f, 0.f, 0.f, 0.f};
  float ui[12] = {0.f, 0.f, 0.f, 0.f, 0.f, 0.f, 0.f, 0.f, 0.f, 0.f, 0.f, 0.f};
  for (int m = 0; m < NM; ++m) {
    float cx = Cx[m * J + j], sx = Sx[m * J + j];
#pragma unroll
    for (int q = 0; q < 16; ++q) {
      float tr = Tr[(size_t)(q * NM + m) * J + j];
      float ti = Ti[(size_t)(q * NM + m) * J + j];
      if (q < 4) ur[q]     += tr * cx - ti * sx;  // real part
      else       ui[q - 4] += tr * sx + ti * cx;  // imag part
    }
  }
  float e = 0.f, f0 = 0.f, f1 = 0.f, f2 = 0.f;
#pragma unroll
  for (int c = 0; c < CCH; ++c) {
    float q = charge[j * CCH + c];
    e  += q * ur[c];
    f0 += q * ui[0 + c];
    f1 += q * ui[4 + c];
    f2 += q * ui[8 + c];
  }
  out[b * J + j] = e / (2.0f * VBOX);
  float* F = out + 2 * J;                 // Force block after energies
  F[(size_t)(b * J + j) * 3 + 0] = f0 / VBOX;
  F[(size_t)(b * J + j) * 3 + 1] = f1 / VBOX;
  F[(size_t)(b * J + j) * 3 + 2] = f2 / VBOX;
}

extern "C" void kernel_launch(void* const* d_in, const int* in_sizes, int n_in,
                              void* d_out, int out_size, void* d_ws, size_t ws_size,
                              hipStream_t stream) {
  const float* pos    = (const float*)d_in[0];
  const float* charge = (const float*)d_in[1];
  const float* shOO = (const float*)d_in[2];
  const float* amOO = (const float*)d_in[3];
  const float* shOH = (const float*)d_in[4];
  const float* amOH = (const float*)d_in[5];
  const float* shHO = (const float*)d_in[6];
  const float* amHO = (const float*)d_in[7];
  const float* shHH = (const float*)d_in[8];
  const float* amHH = (const float*)d_in[9];
  float* out = (float*)d_out;
  float* ws  = (float*)d_ws;

  float *Cx = ws + OFF_CX, *Sx = ws + OFF_SX;
  float *Cy = ws + OFF_CY, *Sy = ws + OFF_SY;
  float *Cz = ws + OFF_CZ, *Sz = ws + OFF_SZ;
  float *Wc = ws + OFF_WC, *Wss = ws + OFF_WS;
  float *Yr = ws + OFF_YR, *Yi = ws + OFF_YI;
  float *rOr = ws + OFF_ROR, *rOi = ws + OFF_ROI;
  float *rHr = ws + OFF_RHR, *rHi = ws + OFF_RHI;
  float *gOr = ws + OFF_GOR, *gOi = ws + OFF_GOI;
  float *gHr = ws + OFF_GHR, *gHi = ws + OFF_GHI;
  float *Tr = ws + OFF_TR, *Ti = ws + OFF_TI;

  // zero pads (Y rows 124..127, W rows 961..991, g rows 496..511 / cols 961..963)
  // — pads are never re-written by any kernel, so every call is deterministic
  hipMemsetAsync(d_ws, 0, WS_TOTAL * sizeof(float), stream);

  for (int b = 0; b < 2; ++b) {
    const float* posb = pos + (size_t)b * J * 3;
    const float* chb  = charge + (size_t)b * J * CCH;

    phase_build<<<(3 * J + 127) / 128, 128, 0, stream>>>(posb, Cx, Sx, Cy, Sy, Cz, Sz);
    w_build<<<(NP * J + 255) / 256, 256, 0, stream>>>(Cy, Sy, Cz, Sz, Wc, Wss);
    y_build<<<(CCH * NM * J + 255) / 256, 256, 0, stream>>>(chb, Cx, Sx, Yr, Yi);

    // nufft1 (sign -1): r = Y * W^T over particles; NT, K-contiguous
    dim3 g1(NPPAD / 32, CMP / 32);
    cgemm_wmma<true, false><<<g1, 32, 0, stream>>>(Yr, Yi, J, Wc, Wss, J,
                                                   rOr, rOi, NPPAD, JOs);
    cgemm_wmma<true, false><<<g1, 32, 0, stream>>>(Yr + JOs, Yi + JOs, J,
                                                   Wc + JOs, Wss + JOs, J,
                                                   rHr, rHi, NPPAD, J - JOs);

    g_build<<<(NM * NP + 255) / 256, 256, 0, stream>>>(
        rOr, rOi, rHr, rHi,
        shOO, amOO, shOH, amOH, shHO, amHO, shHH, amHH,
        gOr, gOi, gHr, gHi);

    // nufft2 stage A (sign +1): T = g * W over mesh (n,p); NN
    dim3 g2o(JOs / 32, QMP / 32), g2h((J - JOs) / 32, QMP / 32);
    cgemm_wmma<false, true><<<g2o, 32, 0, stream>>>(gOr, gOi, KNP, Wc, Wss, J,
                                                    Tr, Ti, J, KNP);
    cgemm_wmma<false, true><<<g2h, 32, 0, stream>>>(gHr, gHi, KNP, Wc + JOs, Wss + JOs, J,
                                                    Tr + JOs, Ti + JOs, J, KNP);

    finalize<<<(J + 127) / 128, 128, 0, stream>>>(Tr, Ti, Cx, Sx, chb, out, b);
  }
}